// kappaLinear_12790412607642
// MI455X (gfx1250) — compile-verified
//
#include <hip/hip_runtime.h>
#include <math.h>
#include <stdint.h>

typedef __attribute__((ext_vector_type(2))) float v2f;
typedef __attribute__((ext_vector_type(8))) float v8f;

#define EPSC     1e-15f
#define CLIPC    0.9999999f      /* 1 - 1e-7 */
#define MAXNORM  0.996f          /* 1 - 4e-3 */

#define BM 128
#define BN 256
#define BK 16
#define LDSS (BK + 4)            /* padded LDS row stride (floats) */

// ---------------------------------------------------------------------------
// CDNA5 async global->LDS copy (16B per lane), tracked by ASYNCcnt.
// VDST = LDS byte offset VGPR, VADDR = 64-bit global address, saddr = off.
// ---------------------------------------------------------------------------
__device__ __forceinline__ uint32_t lds_off(const void* p) {
    // generic LDS pointer = {SHARED_BASE[63:32], lds_byte_offset[31:0]}
    return (uint32_t)(uintptr_t)p;
}

__device__ __forceinline__ void async_cp_b128(uint32_t lds, const float* g) {
    asm volatile("global_load_async_to_lds_b128 %0, %1, off"
                 :: "v"(lds), "v"(g) : "memory");
}

__device__ __forceinline__ void wait_async0() {
    asm volatile("s_wait_asynccnt 0" ::: "memory");
}

// ---------------------------------------------------------------------------
// GEMM: MX[m,n] = sum_k X[m,k] * W[n,k]   (fp32 WMMA 16x16x4)
// 256 threads = 8 waves (2x4); block tile 128x256; wave tile 64x64 (4x4).
// Double-buffered LDS fed by async copies.
// ---------------------------------------------------------------------------
__global__ __launch_bounds__(256) void wmma_gemm_xwt(
    const float* __restrict__ X, const float* __restrict__ W,
    float* __restrict__ MX, int M, int N, int K)
{
    __shared__ float As[2][BM * LDSS];
    __shared__ float Bs[2][BN * LDSS];

    const int tid  = threadIdx.x;
    const int wave = tid >> 5;       // 0..7
    const int lane = tid & 31;
    const int wm   = wave >> 2;      // 0..1 : 64-row group
    const int wn   = wave & 3;       // 0..3 : 64-col group
    const int bm   = blockIdx.y * BM;
    const int bn   = blockIdx.x * BN;

    v8f acc[4][4] = {};

    const int mrow = lane & 15;            // M (or N) index within 16-tile
    const int koff = (lane >> 4) << 1;     // 0 for lanes 0-15, 2 for 16-31

    // Stage one K-chunk (A: 128x16, B: 256x16) into LDS buffer `buf`.
    auto stage = [&](int buf, int kb) {
        #pragma unroll
        for (int i = 0; i < 2; ++i) {           // A: 512 float4 / 256 thr
            int q   = tid * 2 + i;
            int row = q >> 2;
            int c4  = (q & 3) << 2;
            async_cp_b128(lds_off(&As[buf][row * LDSS + c4]),
                          X + (size_t)(bm + row) * K + kb + c4);
        }
        #pragma unroll
        for (int i = 0; i < 4; ++i) {           // B: 1024 float4 / 256 thr
            int q   = tid * 4 + i;
            int row = q >> 2;
            int c4  = (q & 3) << 2;
            async_cp_b128(lds_off(&Bs[buf][row * LDSS + c4]),
                          W + (size_t)(bn + row) * K + kb + c4);
        }
    };

    stage(0, 0);
    wait_async0();
    __syncthreads();

    int buf = 0;
    for (int kb = 0; kb < K; kb += BK) {
        if (kb + BK < K) stage(buf ^ 1, kb + BK);   // overlap with compute

        const float* Ab = &As[buf][0];
        const float* Bb = &Bs[buf][0];
        #pragma unroll
        for (int kk = 0; kk < BK; kk += 4) {
            v2f afrag[4], bfrag[4];
            #pragma unroll
            for (int tm = 0; tm < 4; ++tm) {
                const float* p = &Ab[(wm * 64 + tm * 16 + mrow) * LDSS + kk + koff];
                v2f a; a.x = p[0]; a.y = p[1];
                afrag[tm] = a;
            }
            #pragma unroll
            for (int tn = 0; tn < 4; ++tn) {
                const float* p = &Bb[(wn * 64 + tn * 16 + mrow) * LDSS + kk + koff];
                v2f b; b.x = p[0]; b.y = p[1];
                bfrag[tn] = b;
            }
            #pragma unroll
            for (int tm = 0; tm < 4; ++tm)
                #pragma unroll
                for (int tn = 0; tn < 4; ++tn)
                    acc[tm][tn] = __builtin_amdgcn_wmma_f32_16x16x4_f32(
                        false, afrag[tm], false, bfrag[tn],
                        (short)0, acc[tm][tn], false, false);
        }

        wait_async0();        // our async copies into buf^1 landed in LDS
        __syncthreads();      // everyone done copying + done reading buf
        buf ^= 1;
    }

    // Store D tiles. 32-bit C/D layout: VGPR j -> M = j + (lane>=16 ? 8 : 0),
    // N = lane & 15.
    const int nl    = lane & 15;
    const int mhalf = (lane >> 4) << 3;   // 0 or 8
    #pragma unroll
    for (int tm = 0; tm < 4; ++tm) {
        #pragma unroll
        for (int tn = 0; tn < 4; ++tn) {
            size_t base_m = (size_t)(bm + wm * 64 + tm * 16 + mhalf);
            int    col    = bn + wn * 64 + tn * 16 + nl;
            #pragma unroll
            for (int j = 0; j < 8; ++j)
                MX[(base_m + j) * N + col] = acc[tm][tn][j];
        }
    }
}

// ---------------------------------------------------------------------------
// Per-row sum of squares of X (for x_norm).
// ---------------------------------------------------------------------------
__global__ __launch_bounds__(256) void row_sumsq(
    const float* __restrict__ X, float* __restrict__ out, int K)
{
    __shared__ float red[256];
    int r = blockIdx.x;
    const float* row = X + (size_t)r * K;
    float s = 0.f;
    for (int c = threadIdx.x; c < K; c += 256) { float v = row[c]; s += v * v; }
    red[threadIdx.x] = s;
    __syncthreads();
    for (int off = 128; off > 0; off >>= 1) {
        if (threadIdx.x < off) red[threadIdx.x] += red[threadIdx.x + off];
        __syncthreads();
    }
    if (threadIdx.x == 0) out[r] = red[0];
}

// ---------------------------------------------------------------------------
// kb = project(expmap0(bias)); y2 = ||kb||^2.  Single block.
// ---------------------------------------------------------------------------
__global__ __launch_bounds__(256) void bias_expmap0(
    const float* __restrict__ bias, float* __restrict__ kb,
    float* __restrict__ y2, int N)
{
    __shared__ float red[256];
    __shared__ float sscale;
    float s = 0.f;
    for (int c = threadIdx.x; c < N; c += 256) { float v = bias[c]; s += v * v; }
    red[threadIdx.x] = s;
    __syncthreads();
    for (int off = 128; off > 0; off >>= 1) {
        if (threadIdx.x < off) red[threadIdx.x] += red[threadIdx.x + off];
        __syncthreads();
    }
    if (threadIdx.x == 0) {
        float n  = fmaxf(sqrtf(red[0]), EPSC);
        float th = tanhf(n);
        float scale = th / n;                 // expmap0 scale (c = 1)
        if (th > MAXNORM) scale *= MAXNORM / th;  // projection
        sscale = scale;
        float nn = scale * n;
        *y2 = nn * nn;
    }
    __syncthreads();
    float scale = sscale;
    for (int c = threadIdx.x; c < N; c += 256) kb[c] = bias[c] * scale;
}

// ---------------------------------------------------------------------------
// Per-row: s2 = ||mx_r||^2, t = <mx_r, kb>; fold all hyperbolic scalar math
// into gamma_r, delta_r so that out_r = gamma_r*mx_r + delta_r*kb.
// ---------------------------------------------------------------------------
__global__ __launch_bounds__(256) void row_coeffs(
    const float* __restrict__ MX, const float* __restrict__ kb,
    const float* __restrict__ xn2, const float* __restrict__ y2p,
    float* __restrict__ gamma, float* __restrict__ delta, int N)
{
    __shared__ float red[512];
    int r = blockIdx.x;
    const float* row = MX + (size_t)r * N;
    float s2 = 0.f, t = 0.f;
    for (int c = threadIdx.x; c < N; c += 256) {
        float v = row[c];
        s2 += v * v;
        t  += v * kb[c];
    }
    red[threadIdx.x]       = s2;
    red[256 + threadIdx.x] = t;
    __syncthreads();
    for (int off = 128; off > 0; off >>= 1) {
        if (threadIdx.x < off) {
            red[threadIdx.x]       += red[threadIdx.x + off];
            red[256 + threadIdx.x] += red[256 + threadIdx.x + off];
        }
        __syncthreads();
    }
    if (threadIdx.x == 0) {
        s2 = red[0];
        t  = red[256];
        float y2  = *y2p;
        float xn  = fmaxf(sqrtf(xn2[r]), EPSC);
        float mxn = fmaxf(sqrtf(s2), EPSC);
        // mobius_matvec: tanh(|Mx|/|x| * artanh(|x|)) * Mx / |Mx|
        float a   = fminf(xn, CLIPC);
        float art = 0.5f * logf((1.f + a) / (1.f - a));
        float th  = tanhf(mxn / xn * art);
        float alpha = th / mxn;
        float rn = fabsf(th);
        if (rn > MAXNORM) alpha *= MAXNORM / rn;   // _project(res)
        // mobius_add(alpha*mx, kb) with c = 1:
        float x2 = alpha * alpha * s2;
        float xy = alpha * t;
        float A  = 1.f + 2.f * xy + y2;
        float B  = 1.f - x2;
        float den = fmaxf(1.f + 2.f * xy + x2 * y2, EPSC);
        float g = A * alpha / den;
        float d = B / den;
        // final _project on g*mx + d*kb
        float n2 = g * g * s2 + 2.f * g * d * t + d * d * y2;
        float n  = fmaxf(sqrtf(n2), EPSC);
        if (n > MAXNORM) { float f = MAXNORM / n; g *= f; d *= f; }
        gamma[r] = g;
        delta[r] = d;
    }
}

// ---------------------------------------------------------------------------
// out[r,c] = gamma[r]*MX[r,c] + delta[r]*kb[c]
// ---------------------------------------------------------------------------
__global__ __launch_bounds__(256) void finalize_rows(
    const float* __restrict__ MX, const float* __restrict__ kb,
    const float* __restrict__ gamma, const float* __restrict__ delta,
    float* __restrict__ out, int N)
{
    int r = blockIdx.y;
    int c = blockIdx.x * 256 + threadIdx.x;
    if (c < N) {
        size_t i = (size_t)r * N + c;
        out[i] = gamma[r] * MX[i] + delta[r] * kb[c];
    }
}

// ---------------------------------------------------------------------------
extern "C" void kernel_launch(void* const* d_in, const int* in_sizes, int n_in,
                              void* d_out, int out_size, void* d_ws, size_t ws_size,
                              hipStream_t stream)
{
    const float* x    = (const float*)d_in[0];
    const float* w    = (const float*)d_in[1];
    const float* bias = (const float*)d_in[2];
    float* out = (float*)d_out;

    const int N = in_sizes[2];               // out_dim = 4096
    const int K = in_sizes[1] / N;           // in_dim  = 4096
    const int M = in_sizes[0] / K;           // rows    = 8192

    float* ws    = (float*)d_ws;
    float* mx    = ws;                        // M*N
    float* xn2   = mx + (size_t)M * N;        // M
    float* kb    = xn2 + M;                   // N
    float* y2    = kb + N;                    // 1
    float* gamma = y2 + 1;                    // M
    float* delta = gamma + M;                 // M

    dim3 gemm_grid(N / BN, M / BM);
    wmma_gemm_xwt<<<gemm_grid, 256, 0, stream>>>(x, w, mx, M, N, K);

    row_sumsq<<<M, 256, 0, stream>>>(x, xn2, K);
    bias_expmap0<<<1, 256, 0, stream>>>(bias, kb, y2, N);
    row_coeffs<<<M, 256, 0, stream>>>(mx, kb, xn2, y2, gamma, delta, N);

    dim3 fin_grid((N + 255) / 256, M);
    finalize_rows<<<fin_grid, 256, 0, stream>>>(mx, kb, gamma, delta, out, N);
}